// CognitiveGNN_25829933318736
// MI455X (gfx1250) — compile-verified
//
#include <hip/hip_runtime.h>
#include <hip/hip_bf16.h>
#include <math.h>

// ---------------- problem constants (match reference) ----------------
constexpr int NN   = 50000;            // nodes (50000 = 16 * 3125: exact 16-row tiling)
constexpr int NE   = 800000;           // edges
constexpr int NEN  = NE + NN;          // edges + self loops
constexpr int IND  = 128;              // input dim
constexpr int HID  = 64;
constexpr int HEADS = 4;
constexpr int HH   = HEADS * HID;      // 256
constexpr int NG   = 64;               // graphs
constexpr int ODIM = 16;

static_assert(NN % 16 == 0, "M tiles must be exact");

typedef float v2f __attribute__((ext_vector_type(2)));
typedef float v8f __attribute__((ext_vector_type(8)));

// ---------------- small helpers ----------------
__device__ __forceinline__ unsigned encf(float f) {
    unsigned u = __float_as_uint(f);
    return (u & 0x80000000u) ? ~u : (u | 0x80000000u);
}
__device__ __forceinline__ float decf(unsigned e) {
    unsigned u = (e & 0x80000000u) ? (e & 0x7FFFFFFFu) : ~e;
    return __uint_as_float(u);
}
#define ENC_NEG_INF 0x007FFFFFu   // encf(-inf)

__device__ __forceinline__ void edge_sd(const int* ei, int e, int& s, int& d) {
    if (e < NE) { s = ei[e]; d = ei[NE + e]; }
    else        { s = d = e - NE; }
}
__device__ __forceinline__ float lrelu02(float x) { return x > 0.f ? x : 0.2f * x; }

// ---------------- fill kernels ----------------
__global__ void k_fill_f32(float* p, float v, long n) {
    long i = (long)blockIdx.x * blockDim.x + threadIdx.x;
    if (i < n) p[i] = v;
}
__global__ void k_fill_u32(unsigned* p, unsigned v, long n) {
    long i = (long)blockIdx.x * blockDim.x + threadIdx.x;
    if (i < n) p[i] = v;
}

// ---------------- WMMA fp32 GEMM ----------------
// One wave computes a 16 x (16*NT) tile of C = A[MxK] @ B[KxN] (row-major).
// M must be a multiple of 16 (true for all call sites: M == NN).
// A 16x4 tile : lane 0-15 hold rows 0..15; per lane float2 = (k+kb, k+kb+1),
//               kb = 2*(lane>>4). Adjacent in memory -> one b64 load.
// B 4x16 tile : lane 0-15 hold cols; same k striping.
// D 16x16     : element (v + 8*(lane>>4), lane&15) in vgpr v.
template<int K, int NT>
__device__ __forceinline__ void wmma_accum_tile(const float* __restrict__ A,
                                                const float* __restrict__ B,
                                                int N, int m0, int n0, v8f (&acc)[NT]) {
    const int lane = threadIdx.x & 31;
    const int lm   = lane & 15;
    const int kb   = (lane >> 4) * 2;
    const long arow = (long)(m0 + lm) * K;
    #pragma unroll 4
    for (int k0 = 0; k0 < K; k0 += 4) {
        v2f a = *(const v2f*)(A + arow + k0 + kb);      // global_load_b64
        const float* b0 = B + (long)(k0 + kb) * N;
        const float* b1 = B + (long)(k0 + kb + 1) * N;
        #pragma unroll
        for (int t = 0; t < NT; ++t) {
            v2f b;
            int col = n0 + 16 * t + lm;
            b.x = b0[col];
            b.y = b1[col];
            acc[t] = __builtin_amdgcn_wmma_f32_16x16x4_f32(false, a, false, b,
                                                           (short)0, acc[t], false, false);
        }
    }
}

template<int NT>
__device__ __forceinline__ void wmma_store_tile(float* __restrict__ C, const v8f (&acc)[NT],
                                                int N, int m0, int n0,
                                                const float* bias, int do_relu) {
    const int lane = threadIdx.x & 31;
    const int rb   = m0 + 8 * (lane >> 4);
    #pragma unroll
    for (int t = 0; t < NT; ++t) {
        const int col = n0 + 16 * t + (lane & 15);
        float badd = bias ? bias[col] : 0.f;
        #pragma unroll
        for (int v = 0; v < 8; ++v) {
            float val = acc[t][v] + badd;
            if (do_relu) val = fmaxf(val, 0.f);
            C[(long)(rb + v) * N + col] = val;
        }
    }
}

template<int K, int NT>
__global__ void k_gemm(const float* __restrict__ A, const float* __restrict__ B,
                       const float* __restrict__ bias, float* __restrict__ C,
                       int N, int do_relu) {
    int m0 = blockIdx.x * 16, n0 = blockIdx.y * 16 * NT;
    v8f acc[NT] = {};
    wmma_accum_tile<K, NT>(A, B, N, m0, n0, acc);
    wmma_store_tile<NT>(C, acc, N, m0, n0, bias, do_relu);
}

// C = relu(A1@B1 + A2@B2 + bias)   (SAGE fused layer)
template<int K, int NT>
__global__ void k_gemm2(const float* __restrict__ A1, const float* __restrict__ B1,
                        const float* __restrict__ A2, const float* __restrict__ B2,
                        const float* __restrict__ bias, float* __restrict__ C, int N) {
    int m0 = blockIdx.x * 16, n0 = blockIdx.y * 16 * NT;
    v8f acc[NT] = {};
    wmma_accum_tile<K, NT>(A1, B1, N, m0, n0, acc);
    wmma_accum_tile<K, NT>(A2, B2, N, m0, n0, acc);
    wmma_store_tile<NT>(C, acc, N, m0, n0, bias, 1);
}

// ---------------- GAT attention pieces ----------------
// a_src[n][h] = sum_d h1[n, h*64+d] * att[h][d]
__global__ void k_att(const float* __restrict__ h1,
                      const float* __restrict__ att_src, const float* __restrict__ att_dst,
                      float* __restrict__ asrc, float* __restrict__ adst) {
    int idx = blockIdx.x * blockDim.x + threadIdx.x;
    if (idx >= NN * HEADS) return;
    int n = idx >> 2, h = idx & 3;
    const float* hp = h1 + (long)n * HH + h * HID;
    const float* as = att_src + h * HID;
    const float* ad = att_dst + h * HID;
    float s0 = 0.f, s1 = 0.f;
    for (int d = 0; d < HID; ++d) { float v = hp[d]; s0 += v * as[d]; s1 += v * ad[d]; }
    asrc[idx] = s0; adst[idx] = s1;
}

__global__ void k_logit_max(const int* __restrict__ ei,
                            const float* __restrict__ asrc, const float* __restrict__ adst,
                            unsigned* __restrict__ mmax) {
    int e = blockIdx.x * blockDim.x + threadIdx.x;
    if (e >= NEN) return;
    int s, d; edge_sd(ei, e, s, d);
    #pragma unroll
    for (int h = 0; h < HEADS; ++h) {
        float lg = lrelu02(asrc[s * HEADS + h] + adst[d * HEADS + h]);
        atomicMax(&mmax[d * HEADS + h], encf(lg));
    }
}

__global__ void k_expsum(const int* __restrict__ ei,
                         const float* __restrict__ asrc, const float* __restrict__ adst,
                         const unsigned* __restrict__ mmax,
                         float* __restrict__ ealpha, float* __restrict__ ssum) {
    int e = blockIdx.x * blockDim.x + threadIdx.x;
    if (e >= NEN) return;
    int s, d; edge_sd(ei, e, s, d);
    #pragma unroll
    for (int h = 0; h < HEADS; ++h) {
        float lg = lrelu02(asrc[s * HEADS + h] + adst[d * HEADS + h]);
        float m  = decf(mmax[d * HEADS + h]);
        float ev = __expf(lg - m);
        ealpha[(long)e * HEADS + h] = ev;
        atomicAdd(&ssum[d * HEADS + h], ev);
    }
}

// block = 256 threads = one edge's full [4 heads x 64 dims] message
__global__ void k_gat_scatter(const int* __restrict__ ei, const float* __restrict__ h1,
                              const float* __restrict__ ealpha, const float* __restrict__ ssum,
                              float* __restrict__ out) {
    int e = blockIdx.x;
    int t = threadIdx.x;          // t = h*64 + d
    int h = t >> 6;
    int s, d; edge_sd(ei, e, s, d);
    float alpha = ealpha[(long)e * HEADS + h] / (ssum[d * HEADS + h] + 1e-16f);
    atomicAdd(&out[(long)d * HH + t], h1[(long)s * HH + t] * alpha);
}

__global__ void k_bias_relu(float* __restrict__ p, const float* __restrict__ bias, long n, int c) {
    long i = (long)blockIdx.x * blockDim.x + threadIdx.x;
    if (i < n) p[i] = fmaxf(p[i] + bias[i % c], 0.f);
}

// ---------------- SAGE ----------------
__global__ void k_sage_gather(const int* __restrict__ ei, const float* __restrict__ h2,
                              float* __restrict__ ssum, float* __restrict__ cnt) {
    int e = blockIdx.x;           // 0..NE (no self loops)
    int t = threadIdx.x;
    int s = ei[e], d = ei[NE + e];
    atomicAdd(&ssum[(long)d * HH + t], h2[(long)s * HH + t]);
    if (t == 0) atomicAdd(&cnt[d], 1.f);
}

__global__ void k_sage_mean(float* __restrict__ ssum, const float* __restrict__ cnt) {
    long i = (long)blockIdx.x * blockDim.x + threadIdx.x;
    if (i < (long)NN * HH) ssum[i] /= fmaxf(cnt[i >> 8], 1.f);
}

// ---------------- GCN ----------------
__global__ void k_deg(const int* __restrict__ ei, float* __restrict__ deg) {
    int e = blockIdx.x * blockDim.x + threadIdx.x;
    if (e >= NEN) return;
    int s, d; edge_sd(ei, e, s, d);
    atomicAdd(&deg[d], 1.f);
}
__global__ void k_dinv(float* __restrict__ deg) {
    int i = blockIdx.x * blockDim.x + threadIdx.x;
    if (i < NN) { float v = deg[i]; deg[i] = v > 0.f ? rsqrtf(v) : 0.f; }
}
__global__ void k_gcn_scatter(const int* __restrict__ ei, const float* __restrict__ hlin,
                              const float* __restrict__ dinv, float* __restrict__ out) {
    int e = blockIdx.x;
    int t = threadIdx.x;          // 0..63
    int s, d; edge_sd(ei, e, s, d);
    float norm = dinv[s] * dinv[d];
    atomicAdd(&out[(long)d * HID + t], hlin[(long)s * HID + t] * norm);
}

// ---------------- pooling + classifier ----------------
__global__ void k_pool(const float* __restrict__ out2, const float* __restrict__ gcn_b,
                       const int* __restrict__ batch,
                       float* __restrict__ pool, float* __restrict__ gcnt) {
    long idx = (long)blockIdx.x * blockDim.x + threadIdx.x;
    if (idx >= (long)NN * HID) return;
    int n = (int)(idx >> 6), c = (int)(idx & 63);
    int g = batch[n];
    atomicAdd(&pool[g * HID + c], out2[idx] + gcn_b[c]);
    if (c == 0) atomicAdd(&gcnt[g], 1.f);
}

__global__ void k_cls(const float* __restrict__ pool, const float* __restrict__ gcnt,
                      const float* __restrict__ w1, const float* __restrict__ b1,
                      const float* __restrict__ w2, const float* __restrict__ b2,
                      float* __restrict__ out) {
    int g = threadIdx.x;
    if (g >= NG) return;
    float ic = 1.f / fmaxf(gcnt[g], 1.f);
    float z[32];
    for (int j = 0; j < 32; ++j) {
        float s = b1[j];
        for (int c = 0; c < HID; ++c) s += pool[g * HID + c] * ic * w1[c * 32 + j];
        z[j] = fmaxf(s, 0.f);
    }
    for (int o = 0; o < ODIM; ++o) {
        float s = b2[o];
        for (int j = 0; j < 32; ++j) s += z[j] * w2[j * ODIM + o];
        out[g * ODIM + o] = s;
    }
}

// ---------------- driver ----------------
static inline dim3 g1(long n, int b) { return dim3((unsigned)((n + b - 1) / b)); }

extern "C" void kernel_launch(void* const* d_in, const int* in_sizes, int n_in,
                              void* d_out, int out_size, void* d_ws, size_t ws_size,
                              hipStream_t stream) {
    const float* x        = (const float*)d_in[0];
    const int*   ei       = (const int*)d_in[1];
    const int*   batch    = (const int*)d_in[2];
    const float* gat_w    = (const float*)d_in[3];
    const float* gat_as   = (const float*)d_in[4];
    const float* gat_ad   = (const float*)d_in[5];
    const float* gat_b    = (const float*)d_in[6];
    const float* sage_wl  = (const float*)d_in[7];
    const float* sage_bl  = (const float*)d_in[8];
    const float* sage_wr  = (const float*)d_in[9];
    const float* gcn_w    = (const float*)d_in[10];
    const float* gcn_b    = (const float*)d_in[11];
    const float* cls_w1   = (const float*)d_in[12];
    const float* cls_b1   = (const float*)d_in[13];
    const float* cls_w2   = (const float*)d_in[14];
    const float* cls_b2   = (const float*)d_in[15];
    float* out = (float*)d_out;

    // ---- workspace arena (floats) ----
    float* W = (float*)d_ws;
    size_t o = 0;
    float*    h1     = W + o; o += (size_t)NN * HH;        // GAT pre-act h; later SAGE sum/mean
    float*    h2     = W + o; o += (size_t)NN * HH;        // GAT agg/out; later hlin + out2
    float*    esc    = W + o; o += (size_t)NEN * HEADS;    // ealpha; later h3 (NN*HID fits)
    float*    asrc   = W + o; o += (size_t)NN * HEADS;
    float*    adst   = W + o; o += (size_t)NN * HEADS;
    unsigned* mmax   = (unsigned*)(W + o); o += (size_t)NN * HEADS;
    float*    ssum   = W + o; o += (size_t)NN * HEADS;
    float*    cnt    = W + o; o += (size_t)NN;
    float*    deg    = W + o; o += (size_t)NN;
    float*    pool   = W + o; o += (size_t)NG * HID;
    float*    gcnt   = W + o; o += (size_t)NG;
    (void)ws_size; (void)n_in; (void)in_sizes; (void)out_size;

    float* h3   = esc;                     // [NN x 64], reuses ealpha region
    float* hlin = h2;                      // [NN x 64], reuses GAT-out region
    float* out2 = h2 + (size_t)NN * HID;   // [NN x 64], second quarter of GAT-out region

    const int B = 256;
    const int MT = NN / 16;                // 3125 M-tiles, exact

    // ===== GAT =====
    k_fill_f32<<<g1((long)NN * HH, B), B, 0, stream>>>(h2, 0.f, (long)NN * HH);
    k_fill_u32<<<g1((long)NN * HEADS, B), B, 0, stream>>>(mmax, ENC_NEG_INF, (long)NN * HEADS);
    k_fill_f32<<<g1((long)NN * HEADS, B), B, 0, stream>>>(ssum, 0.f, (long)NN * HEADS);

    // h1 = x @ gat_w  (50000x128 @ 128x256), 16x32 tile per wave
    k_gemm<IND, 2><<<dim3(MT, HH / 32), 32, 0, stream>>>(x, gat_w, nullptr, h1, HH, 0);
    k_att<<<g1((long)NN * HEADS, B), B, 0, stream>>>(h1, gat_as, gat_ad, asrc, adst);
    k_logit_max<<<g1(NEN, B), B, 0, stream>>>(ei, asrc, adst, mmax);
    k_expsum<<<g1(NEN, B), B, 0, stream>>>(ei, asrc, adst, mmax, esc, ssum);
    k_gat_scatter<<<dim3(NEN), HH, 0, stream>>>(ei, h1, esc, ssum, h2);
    k_bias_relu<<<g1((long)NN * HH, B), B, 0, stream>>>(h2, gat_b, (long)NN * HH, HH);

    // ===== SAGE =====
    k_fill_f32<<<g1((long)NN * HH, B), B, 0, stream>>>(h1, 0.f, (long)NN * HH);  // reuse as sum
    k_fill_f32<<<g1(NN, B), B, 0, stream>>>(cnt, 0.f, NN);
    k_sage_gather<<<dim3(NE), HH, 0, stream>>>(ei, h2, h1, cnt);
    k_sage_mean<<<g1((long)NN * HH, B), B, 0, stream>>>(h1, cnt);
    // h3 = relu(mean @ wl + h2 @ wr + bl)   (K=256, N=64)
    k_gemm2<HH, 2><<<dim3(MT, HID / 32), 32, 0, stream>>>(h1, sage_wl, h2, sage_wr,
                                                          sage_bl, h3, HID);

    // ===== GCN =====
    k_fill_f32<<<g1(NN, B), B, 0, stream>>>(deg, 0.f, NN);
    k_deg<<<g1(NEN, B), B, 0, stream>>>(ei, deg);
    k_dinv<<<g1(NN, B), B, 0, stream>>>(deg);
    // hlin = h3 @ gcn_w  (K=64, N=64)  -- overwrites first quarter of h2 region (h2 now dead)
    k_gemm<HID, 2><<<dim3(MT, HID / 32), 32, 0, stream>>>(h3, gcn_w, nullptr, hlin, HID, 0);
    k_fill_f32<<<g1((long)NN * HID, B), B, 0, stream>>>(out2, 0.f, (long)NN * HID);
    k_gcn_scatter<<<dim3(NEN), HID, 0, stream>>>(ei, hlin, deg, out2);

    // ===== pool + classifier =====
    k_fill_f32<<<g1(NG * HID, B), B, 0, stream>>>(pool, 0.f, NG * HID);
    k_fill_f32<<<g1(NG, B), B, 0, stream>>>(gcnt, 0.f, NG);
    k_pool<<<g1((long)NN * HID, B), B, 0, stream>>>(out2, gcn_b, batch, pool, gcnt);
    k_cls<<<dim3(1), NG, 0, stream>>>(pool, gcnt, cls_w1, cls_b1, cls_w2, cls_b2, out);
}